// message_13082470383778
// MI455X (gfx1250) — compile-verified
//
#include <hip/hip_runtime.h>

#define E_EDGES 200000
#define N_NODES 12500
#define D_FEAT  128
#define NR_BF   20
#define RCUT_F  5.0f
#define R_TOTAL (E_EDGES * 3)

typedef __attribute__((ext_vector_type(2))) float v2f;
typedef __attribute__((ext_vector_type(8))) float v8f;
typedef __attribute__((ext_vector_type(4))) unsigned int u32x4;
typedef __attribute__((ext_vector_type(8))) int i32x8;
typedef __attribute__((ext_vector_type(4))) int i32x4;

#define LDS_W2_FLOATS (384 * 128)                 /* 192 KB */
#define LDS_H_BASE    LDS_W2_FLOATS
#define LDS_TOTAL_BYTES ((LDS_W2_FLOATS + 8 * 16 * D_FEAT) * 4)  /* 256 KB */

__device__ __forceinline__ v8f wmma4(float a0, float a1, float b0, float b1, v8f c) {
  v2f a = {a0, a1};
  v2f b = {b0, b1};
  return __builtin_amdgcn_wmma_f32_16x16x4_f32(false, a, false, b, (short)0, c, false, false);
}

// ---------------- deterministic global-norm reduction (for org = r/||r||_F) ---
__global__ void reduce_partial_kernel(const float* __restrict__ r, float* __restrict__ part) {
  __shared__ float sm[256];
  int base = blockIdx.x * 4096;
  float acc = 0.0f;
  for (int i = threadIdx.x; i < 4096; i += 256) {
    int g = base + i;
    if (g < R_TOTAL) { float x = r[g]; acc += x * x; }
  }
  sm[threadIdx.x] = acc;
  __syncthreads();
  for (int st = 128; st > 0; st >>= 1) {
    if ((int)threadIdx.x < st) sm[threadIdx.x] += sm[threadIdx.x + st];
    __syncthreads();
  }
  if (threadIdx.x == 0) part[blockIdx.x] = sm[0];
}

__global__ void reduce_final_kernel(const float* __restrict__ part, float* __restrict__ invR, int n) {
  __shared__ float sm[256];
  float acc = 0.0f;
  for (int i = threadIdx.x; i < n; i += 256) acc += part[i];
  sm[threadIdx.x] = acc;
  __syncthreads();
  for (int st = 128; st > 0; st >>= 1) {
    if ((int)threadIdx.x < st) sm[threadIdx.x] += sm[threadIdx.x + st];
    __syncthreads();
  }
  if (threadIdx.x == 0) *invR = 1.0f / sqrtf(sm[0]);
}

// ---------------- main fused edge kernel -------------------------------------
__global__ void __launch_bounds__(256)
pet_edge_kernel(const float* __restrict__ s, const float* __restrict__ r,
                const float* __restrict__ vv, const float* __restrict__ W1,
                const float* __restrict__ b1, const float* __restrict__ W2,
                const float* __restrict__ b2, const float* __restrict__ Ww,
                const float* __restrict__ bw, const int* __restrict__ idx_i,
                const float* __restrict__ invRp,
                float* __restrict__ out_v, float* __restrict__ out_s) {
  extern __shared__ float smem[];
  const int lane = threadIdx.x & 31;
  const int wv   = threadIdx.x >> 5;
  const int hf   = lane >> 4;      // which 16-lane half
  const int m16  = lane & 15;
  const int c    = hf * 2;         // K sub-offset for WMMA f32 16x16x4 A/B layout

  // --- TDM: async-copy W2 (384x128 f32) into the first 192 KB of LDS, one wave issues
#if __has_builtin(__builtin_amdgcn_tensor_load_to_lds)
  if (wv == 0) {
    unsigned long long ga = (unsigned long long)(const void*)W2;
    u32x4 g0;
    g0.x = 1u;                                               // count = 1 valid descriptor
    g0.y = 0u;                                               // lds_addr = 0
    g0.z = (unsigned)(ga & 0xFFFFFFFFu);                     // global_addr[31:0]
    g0.w = (unsigned)((ga >> 32) & 0x01FFFFFFu) | 0x80000000u; // addr[56:32] | type=2
    i32x8 g1;
    g1.s0 = 0x00020000;                 // data_size = 4B
    g1.s1 = (int)(128u << 16);          // tensor_dim0 (lo16 at bit48)
    g1.s2 = (int)(384u << 16);          // tensor_dim1 (lo16 at bit80)
    g1.s3 = (int)(128u << 16);          // tile_dim0 = 128
    g1.s4 = 384;                        // tile_dim1 = 384, tile_dim2 = 0
    g1.s5 = 128;                        // tensor_dim0_stride
    g1.s6 = 0;
    g1.s7 = 0;
    i32x4 gz4 = {0, 0, 0, 0};
    i32x8 gz8 = {0, 0, 0, 0, 0, 0, 0, 0};
    __builtin_amdgcn_tensor_load_to_lds(g0, g1, gz4, gz4, gz8, 0);
    __builtin_amdgcn_s_wait_tensorcnt(0);
  }
#else
  for (int i = threadIdx.x; i < LDS_W2_FLOATS; i += 256) smem[i] = W2[i];
#endif
  __syncthreads();

  const int tile0 = blockIdx.x * 128 + wv * 16;   // this wave's 16-edge tile
  if (tile0 >= E_EDGES) return;

  const float invR = *invRp;

  // per-edge scalars for edge e = tile0 + m16 (A-matrix row owned by this lane)
  int eA = tile0 + m16;
  if (eA >= E_EDGES) eA = E_EDGES - 1;
  float r0 = r[eA * 3 + 0], r1 = r[eA * 3 + 1], r2 = r[eA * 3 + 2];
  float rn   = sqrtf(r0 * r0 + r1 * r1 + r2 * r2);
  float rinv = 1.0f / rn;
  float o0 = r0 * invR, o1 = r1 * invR, o2 = r2 * invR;   // reference uses GLOBAL norm
  int myidx = idx_i[eA];

  // fcut A-fragment (K = 20 -> 5 WMMA k-steps)
  const float PI = 3.14159265358979f;
  float fc[10];
#pragma unroll
  for (int t = 0; t < 5; ++t) {
#pragma unroll
    for (int u = 0; u < 2; ++u) {
      int   kk  = 4 * t + c + u;
      float nv  = (float)(kk + 1);
      float rbf = __sinf(nv * (PI / RCUT_F) * rn) * rinv;
      fc[2 * t + u] = (rbf <= RCUT_F) ? 0.5f * (__cosf(PI * rbf / RCUT_F) + 1.0f) : 0.0f;
    }
  }

  // A-fragment of s (16 edges x 128), reused across all GEMM1 column tiles
  float a[64];
  {
    const float* srow = s + (long long)eA * D_FEAT;
#pragma unroll
    for (int t = 0; t < 32; ++t) {
      float2 p = *(const float2*)(srow + 4 * t + c);
      a[2 * t] = p.x; a[2 * t + 1] = p.y;
    }
  }

  // ---- GEMM1: h = silu(s @ W1^T + b1); store h tile to LDS, one row per edge -----------
  // Two independent accumulator chains (k-steps 0..15 / 16..31) for WMMA ILP.
  float* hbuf = smem + LDS_H_BASE + wv * (16 * D_FEAT);
#pragma unroll 1
  for (int ft = 0; ft < 8; ++ft) {
    int fr = 16 * ft + m16;
    const float* w1row = W1 + fr * D_FEAT;
    v8f acc0 = {0.f, 0.f, 0.f, 0.f, 0.f, 0.f, 0.f, 0.f};
    v8f acc1 = {0.f, 0.f, 0.f, 0.f, 0.f, 0.f, 0.f, 0.f};
#pragma unroll
    for (int t = 0; t < 16; ++t) {
      float2 bA = *(const float2*)(w1row + 4 * t + c);
      float2 bB = *(const float2*)(w1row + 4 * (t + 16) + c);
      acc0 = wmma4(a[2 * t], a[2 * t + 1], bA.x, bA.y, acc0);
      acc1 = wmma4(a[2 * (t + 16)], a[2 * (t + 16) + 1], bB.x, bB.y, acc1);
    }
    float bias = b1[fr];
#pragma unroll
    for (int j = 0; j < 8; ++j) {
      float x  = acc0[j] + acc1[j] + bias;
      float sg = 1.0f / (1.0f + __expf(-x));
      hbuf[(j + 8 * hf) * D_FEAT + fr] = x * sg;     // row = edge, col = feature
    }
  }

  // ---- transpose h to A-layout via LDS --------------------------------------
  float a2[64];
  {
    const float* hrow = hbuf + m16 * D_FEAT;
#pragma unroll
    for (int t = 0; t < 32; ++t) {
      float2 p = *(const float2*)(hrow + 4 * t + c);
      a2[2 * t] = p.x; a2[2 * t + 1] = p.y;
    }
  }

  // ---- GEMM2 (+gate GEMM) + fused epilogue, per 16-wide feature tile --------
#pragma unroll 1
  for (int t = 0; t < 8; ++t) {
    float res[3][8];
#pragma unroll
    for (int part = 0; part < 3; ++part) {
      int f2r = 128 * part + 16 * t + m16;
      // phi tile: h @ W2^T (B from LDS), dual accumulator chains
      v8f acc0 = {0.f, 0.f, 0.f, 0.f, 0.f, 0.f, 0.f, 0.f};
      v8f acc1 = {0.f, 0.f, 0.f, 0.f, 0.f, 0.f, 0.f, 0.f};
      const float* wrow = smem + f2r * D_FEAT;
#pragma unroll
      for (int k = 0; k < 16; ++k) {
        float2 bA = *(const float2*)(wrow + 4 * k + c);
        float2 bB = *(const float2*)(wrow + 4 * (k + 16) + c);
        acc0 = wmma4(a2[2 * k], a2[2 * k + 1], bA.x, bA.y, acc0);
        acc1 = wmma4(a2[2 * (k + 16)], a2[2 * (k + 16) + 1], bB.x, bB.y, acc1);
      }
      float phib = b2[f2r];
      // gate tile: fcut @ Ww^T (K = 20, single chain)
      v8f accw = {0.f, 0.f, 0.f, 0.f, 0.f, 0.f, 0.f, 0.f};
      const float* wwrow = Ww + f2r * NR_BF;
#pragma unroll
      for (int k = 0; k < 5; ++k) {
        float2 b = *(const float2*)(wwrow + 4 * k + c);
        accw = wmma4(fc[2 * k], fc[2 * k + 1], b.x, b.y, accw);
      }
      float bwv = bw[f2r];
#pragma unroll
      for (int j = 0; j < 8; ++j)
        res[part][j] = (accw[j] + bwv) * (acc0[j] + acc1[j] + phib);
    }
    // epilogue: dd = 16*t + m16 (column), rows e = j + 8*hf
    int dd = 16 * t + m16;
#pragma unroll
    for (int j = 0; j < 8; ++j) {
      int   eloc = j + 8 * hf;
      int   idxv = __shfl(myidx, eloc, 32);      // full-EXEC shuffles
      float q0   = __shfl(o0, eloc, 32);
      float q1   = __shfl(o1, eloc, 32);
      float q2   = __shfl(o2, eloc, 32);
      int   eg   = tile0 + eloc;
      if (eg < E_EDGES) {
        float s0v = res[0][j], s1v = res[1][j], s2v = res[2][j];
        atomicAdd(out_s + (long long)idxv * D_FEAT + dd, s1v);
        const float* vrow = vv + (long long)eg * 3 * D_FEAT;
        float* ov = out_v + (long long)idxv * 3 * D_FEAT;
        atomicAdd(ov + 0 * D_FEAT + dd, s2v * q0 + s0v * vrow[0 * D_FEAT + dd]);
        atomicAdd(ov + 1 * D_FEAT + dd, s2v * q1 + s0v * vrow[1 * D_FEAT + dd]);
        atomicAdd(ov + 2 * D_FEAT + dd, s2v * q2 + s0v * vrow[2 * D_FEAT + dd]);
      }
    }
  }
}

extern "C" void kernel_launch(void* const* d_in, const int* in_sizes, int n_in,
                              void* d_out, int out_size, void* d_ws, size_t ws_size,
                              hipStream_t stream) {
  (void)in_sizes; (void)n_in; (void)ws_size;
  const float* s  = (const float*)d_in[0];
  const float* r  = (const float*)d_in[1];
  const float* v  = (const float*)d_in[2];
  const float* W1 = (const float*)d_in[3];
  const float* b1 = (const float*)d_in[4];
  const float* W2 = (const float*)d_in[5];
  const float* b2 = (const float*)d_in[6];
  const float* Ww = (const float*)d_in[7];
  const float* bw = (const float*)d_in[8];
  const int* idx  = (const int*)d_in[9];

  float* out_v = (float*)d_out;
  float* out_s = out_v + (long long)N_NODES * 3 * D_FEAT;
  float* wsf   = (float*)d_ws;           // wsf[0] = 1/||r||_F, wsf[1..] partials

  (void)hipMemsetAsync(d_out, 0, (size_t)out_size * sizeof(float), stream);

  const int nPart = (R_TOTAL + 4095) / 4096;  // 147
  reduce_partial_kernel<<<nPart, 256, 0, stream>>>(r, wsf + 1);
  reduce_final_kernel<<<1, 256, 0, stream>>>(wsf + 1, wsf, nPart);

  (void)hipFuncSetAttribute((const void*)pet_edge_kernel,
                            hipFuncAttributeMaxDynamicSharedMemorySize,
                            LDS_TOTAL_BYTES);
  const int blocks = (E_EDGES + 127) / 128;   // 1563, 8 waves x 16 edges each
  pet_edge_kernel<<<blocks, 256, LDS_TOTAL_BYTES, stream>>>(
      s, r, v, W1, b1, W2, b2, Ww, bw, idx, wsf, out_v, out_s);
}